// LegModel_40561671143581
// MI455X (gfx1250) — compile-verified
//
#include <hip/hip_runtime.h>
#include <cstddef>

typedef float v2f __attribute__((ext_vector_type(2)));
typedef float v8f __attribute__((ext_vector_type(8)));

#define HID 148
#define LD  160   // padded leading dimension (10 x 16 col tiles)

// ---------------------------------------------------------------------------
// Zero-padded weight repack: dst[rows x LD], src[rows x cols]
// ---------------------------------------------------------------------------
__global__ __launch_bounds__(256) void pad_copy_kernel(
    float* __restrict__ dst, const float* __restrict__ src,
    int rows, int cols, int total)
{
  int i = blockIdx.x * blockDim.x + threadIdx.x;
  if (i >= total) return;
  int r = i / LD, c = i - r * LD;
  dst[i] = (c < cols) ? src[(size_t)r * cols + c] : 0.0f;
}

// ---------------------------------------------------------------------------
// fp32 WMMA GEMM: C[M x 160] (+)= A[M x K] @ B[K x 160] (+bias)
// using V_WMMA_F32_16X16X4_F32. One wave computes a 16-row x 80-col slab:
// the A fragment is loaded once per k-step and reused across 5 N tiles
// (5 WMMAs per 11 loads). Two waves cover the 160-col padded row block.
// flags: bit0 = accumulate into C, bit1 = ReLU.
// A-row reads clamped to M-1 (for unpadded device inputs); all C buffers are
// row-padded to multiples of 16 so stores need no guard.
// ---------------------------------------------------------------------------
__global__ __launch_bounds__(256) void wmma_gemm_f32(
    const float* __restrict__ A, int lda, int M,
    const float* __restrict__ B,                  // ldb == LD
    const float* __restrict__ bias, int nbias,
    float* __restrict__ C,                        // ldc == LD
    int K, int Mtiles, int flags)
{
  int wid  = (int)((blockIdx.x * blockDim.x + threadIdx.x) >> 5);
  int lane = (int)(threadIdx.x & 31);
  if (wid >= (Mtiles << 1)) return;              // wave-uniform exit, EXEC stays full
  int mt   = wid >> 1;
  int half = wid & 1;                            // 0: cols 0..79, 1: cols 80..159
  int row0 = mt << 4;
  int colb = half * 80;
  int lo = lane & 15, hi = lane >> 4;

  // A 16x4 fragment: lanes 0-15 hold M=row0+lo K={0,1}; lanes 16-31 K={2,3}
  int ar = row0 + lo; ar = (ar < M) ? ar : (M - 1);
  const float* ap = A + (size_t)ar * lda + (hi << 1);
  // B 4x16 fragments: lanes 0-15 hold N=col K={0,1}; lanes 16-31 K={2,3}
  const float* bp = B + (size_t)(hi << 1) * LD + colb + lo;

  v8f acc[5];
#pragma unroll
  for (int t = 0; t < 5; ++t) acc[t] = (v8f){};

  int ksteps = K >> 2;                           // K is a multiple of 4 (148, 64)
  for (int k = 0; k < ksteps; ++k) {
    v2f a;
    a.x = ap[0]; a.y = ap[1];
#pragma unroll
    for (int t = 0; t < 5; ++t) {
      v2f b;
      b.x = bp[16 * t];
      b.y = bp[LD + 16 * t];
      acc[t] = __builtin_amdgcn_wmma_f32_16x16x4_f32(false, a, false, b,
                                                     (short)0, acc[t],
                                                     false, false);
    }
    ap += 4;
    bp += (size_t)LD << 2;
  }

  // D layout per tile: VGPR r -> row0 + r + 8*hi, col = colb + 16*t + lo
#pragma unroll
  for (int t = 0; t < 5; ++t) {
    int col = colb + 16 * t + lo;
    float bv = 0.0f;
    if (bias) bv = (col < nbias) ? bias[col] : 0.0f;
#pragma unroll
    for (int r = 0; r < 8; ++r) {
      size_t idx = (size_t)(row0 + r + (hi << 3)) * LD + col;
      float v = acc[t][r] + bv;
      if (flags & 1) v += C[idx];
      if (flags & 2) v = fmaxf(v, 0.0f);
      C[idx] = v;
    }
  }
}

// ---------------------------------------------------------------------------
// Edge scatter: one wave32 per edge; coalesced gather of src row, fp32 L2
// atomics into agg[dst], lane 0 bumps the count. ei = [2,E] (src row, dst row).
// ---------------------------------------------------------------------------
__global__ __launch_bounds__(256) void scatter_kernel(
    const float* __restrict__ H, const int* __restrict__ ei, int E,
    float* __restrict__ agg, float* __restrict__ cnt)
{
  int nw   = (int)((gridDim.x * blockDim.x) >> 5);
  int lane = (int)(threadIdx.x & 31);
  for (int e = (int)((blockIdx.x * blockDim.x + threadIdx.x) >> 5); e < E; e += nw) {
    int s = ei[e];
    int d = ei[E + e];
    const float* hs = H + (size_t)s * LD;
    float* ad = agg + (size_t)d * LD;
#pragma unroll
    for (int i = 0; i < 5; ++i) {
      int j = lane + 32 * i;
      if (j < HID) atomicAdd(&ad[j], hs[j]);
    }
    if (lane == 0) atomicAdd(&cnt[d], 1.0f);
  }
}

__global__ __launch_bounds__(256) void mean_kernel(
    float* __restrict__ agg, const float* __restrict__ cnt, int total)
{
  int i = blockIdx.x * blockDim.x + threadIdx.x;
  if (i >= total) return;
  int r = i / LD;
  agg[i] = agg[i] / fmaxf(cnt[r], 1.0f);
}

// ---------------------------------------------------------------------------
// Fused LayerNorm + ReLU over HID=148 cols, one wave32 per row (in place).
// ---------------------------------------------------------------------------
__global__ __launch_bounds__(256) void ln_relu_kernel(
    float* __restrict__ X, int M,
    const float* __restrict__ g, const float* __restrict__ b)
{
  int row  = (int)((blockIdx.x * blockDim.x + threadIdx.x) >> 5);
  int lane = (int)(threadIdx.x & 31);
  if (row >= M) return;
  float* xr = X + (size_t)row * LD;
  float x[5];
  float s = 0.0f;
#pragma unroll
  for (int i = 0; i < 5; ++i) {
    int j = lane + 32 * i;
    x[i] = (j < HID) ? xr[j] : 0.0f;
    s += x[i];
  }
#pragma unroll
  for (int o = 16; o; o >>= 1) s += __shfl_xor(s, o, 32);
  float mean = s * (1.0f / (float)HID);
  float vs = 0.0f;
#pragma unroll
  for (int i = 0; i < 5; ++i) {
    int j = lane + 32 * i;
    float d = (j < HID) ? (x[i] - mean) : 0.0f;
    vs += d * d;
  }
#pragma unroll
  for (int o = 16; o; o >>= 1) vs += __shfl_xor(vs, o, 32);
  float rstd = rsqrtf(vs * (1.0f / (float)HID) + 1e-5f);
#pragma unroll
  for (int i = 0; i < 5; ++i) {
    int j = lane + 32 * i;
    if (j < HID) {
      float v = (x[i] - mean) * rstd * g[j] + b[j];
      xr[j] = fmaxf(v, 0.0f);
    }
  }
}

// Compact copy: padded [M x LD] -> packed [M x HID]
__global__ __launch_bounds__(256) void copy_out_kernel(
    const float* __restrict__ src, float* __restrict__ dst, int total)
{
  int i = blockIdx.x * blockDim.x + threadIdx.x;
  if (i >= total) return;
  int r = i / HID, c = i - r * HID;
  dst[i] = src[(size_t)r * LD + c];
}

// stance head: tanh((h @ W(148x66) + b) / 3)
__global__ __launch_bounds__(256) void stance_kernel(
    const float* __restrict__ h, const float* __restrict__ W,
    const float* __restrict__ b, float* __restrict__ out, int M)
{
  int i = blockIdx.x * blockDim.x + threadIdx.x;
  if (i >= M * 66) return;
  int r = i / 66, t = i - r * 66;
  const float* hr = h + (size_t)r * LD;
  float acc = b[t];
  for (int k = 0; k < HID; ++k) acc = fmaf(hr[k], W[k * 66 + t], acc);
  out[i] = tanhf(acc * (1.0f / 3.0f));
}

// infl tail: softplus(relu_tmp @ W2(148x1) + b2)
__global__ __launch_bounds__(256) void infl_kernel(
    const float* __restrict__ t, const float* __restrict__ W2,
    const float* __restrict__ b2, float* __restrict__ out, int M)
{
  int r = blockIdx.x * blockDim.x + threadIdx.x;
  if (r >= M) return;
  const float* tr = t + (size_t)r * LD;
  float acc = b2[0];
  for (int k = 0; k < HID; ++k) acc = fmaf(tr[k], W2[k], acc);
  out[r] = fmaxf(acc, 0.0f) + log1pf(expf(-fabsf(acc)));   // stable softplus
}

// ---------------------------------------------------------------------------
// Host orchestration
// ---------------------------------------------------------------------------
extern "C" void kernel_launch(void* const* d_in, const int* in_sizes, int n_in,
                              void* d_out, int out_size, void* d_ws, size_t ws_size,
                              hipStream_t stream)
{
  (void)n_in; (void)out_size; (void)ws_size;
  // node types in NODE_COUNTS insertion order:
  // 0 bill, 1 bill_version, 2 legislator, 3 legislator_term, 4 committee, 5 donor, 6 lobby_firm
  const int Mn[7] = {50000, 150000, 2000, 12000, 500, 200000, 5000};
  const int Mp[7] = {50000, 150000, 2000, 12000, 512, 200000, 5008}; // 16-row padded
  const int alphaPos[7] = {0, 1, 4, 5, 2, 3, 6};  // alphabetical position of each type
  // edges in EDGE_META insertion order (d_in[7+j]):
  const int esrc[10] = {1, 1, 2, 3, 3, 6, 6, 5, 3, 1};
  const int edst[10] = {0, 1, 3, 1, 4, 3, 4, 3, 1, 4};
  const int convAlpha[10] = {1, 5, 7, 9, 4, 3, 2, 0, 8, 6}; // alpha position of edge key
  // relations active per layer (src & dst still present in h):
  const int nAct[3] = {10, 6, 3};
  const int actE[3][10] = {
      {0, 1, 2, 3, 4, 5, 6, 7, 8, 9},
      {0, 1, 3, 4, 8, 9, -1, -1, -1, -1},
      {0, 1, 9, -1, -1, -1, -1, -1, -1, -1}};
  const int nDst[3] = {4, 3, 3};
  const int actD[3][4] = {{0, 1, 3, 4}, {0, 1, 4, -1}, {0, 1, 4, -1}};

  auto P = [&](int i) -> const float* { return (const float*)d_in[i]; };

  // ---- workspace bump allocator (fp32) ----
  size_t woff = 0;
  auto alloc = [&](size_t n) -> float* {
    float* p = (float*)d_ws + woff; woff += n; return p;
  };
  float* hbuf[7];
  for (int t = 0; t < 7; ++t) hbuf[t] = alloc((size_t)Mp[t] * LD);
  float* obuf[7] = {nullptr, nullptr, nullptr, nullptr, nullptr, nullptr, nullptr};
  {
    const int dt[4] = {0, 1, 3, 4};
    for (int k = 0; k < 4; ++k) obuf[dt[k]] = alloc((size_t)Mp[dt[k]] * LD);
  }
  float* agg = alloc((size_t)150000 * LD);
  float* cnt = alloc(150016);
  float* encWp[7];
  for (int t = 0; t < 7; ++t) encWp[t] = alloc((size_t)64 * LD);
  float* Wlp[3][10] = {};
  float* Wrp[3][10] = {};
  for (int L = 0; L < 3; ++L)
    for (int a = 0; a < nAct[L]; ++a) {
      int j = actE[L][a];
      Wlp[L][j] = alloc((size_t)HID * LD);
      Wrp[L][j] = alloc((size_t)HID * LD);
    }
  float* W1p = alloc((size_t)HID * LD);
  float* tmp = alloc((size_t)512 * LD);

  // ---- repack weights into zero-padded [K x LD] buffers ----
  auto pad = [&](float* dst, const float* src, int rows, int cols) {
    int tot = rows * LD;
    pad_copy_kernel<<<(tot + 255) / 256, 256, 0, stream>>>(dst, src, rows, cols, tot);
  };
  for (int t = 0; t < 7; ++t) pad(encWp[t], P(17 + 4 * alphaPos[t] + 0), 64, HID);
  for (int L = 0; L < 3; ++L)
    for (int a = 0; a < nAct[L]; ++a) {
      int j = actE[L][a];
      int cb = 61 + 44 * L + 3 * convAlpha[j];   // Wl, Wr, bl
      pad(Wlp[L][j], P(cb + 0), HID, HID);
      pad(Wrp[L][j], P(cb + 1), HID, HID);
    }
  pad(W1p, P(45), HID, HID);                      // infl['committee'].W1

  auto gemm = [&](const float* A, int lda, int M, int Mpad, const float* B,
                  const float* bias, int nbias, float* C, int K, int flags) {
    int Mtiles = Mpad >> 4;
    int waves = Mtiles << 1;                     // 2 waves per 16-row block
    wmma_gemm_f32<<<(waves + 7) / 8, 256, 0, stream>>>(
        A, lda, M, B, bias, nbias, C, K, Mtiles, flags);
  };

  // ---- encoder: h[t] = relu(ln(x @ W + b; g, be)) ----
  for (int t = 0; t < 7; ++t) {
    int eb = 17 + 4 * alphaPos[t];               // W, b, be, g
    gemm(P(t), 64, Mn[t], Mp[t], encWp[t], P(eb + 1), HID, hbuf[t], 64, 0);
    ln_relu_kernel<<<(Mn[t] + 7) / 8, 256, 0, stream>>>(hbuf[t], Mn[t],
                                                        P(eb + 3), P(eb + 2));
  }

  // ---- 3 HeteroSAGE layers ----
  for (int L = 0; L < 3; ++L) {
    for (int k = 0; k < nDst[L]; ++k) {
      int d = actD[L][k];
      hipMemsetAsync(obuf[d], 0, (size_t)Mp[d] * LD * sizeof(float), stream);
    }
    for (int a = 0; a < nAct[L]; ++a) {
      int j = actE[L][a];
      int s = esrc[j], d = edst[j];
      int E = in_sizes[7 + j] / 2;
      hipMemsetAsync(agg, 0, (size_t)Mp[d] * LD * sizeof(float), stream);
      hipMemsetAsync(cnt, 0, (size_t)Mp[d] * sizeof(float), stream);
      scatter_kernel<<<(E + 7) / 8, 256, 0, stream>>>(
          hbuf[s], (const int*)d_in[7 + j], E, agg, cnt);
      int tot = Mp[d] * LD;
      mean_kernel<<<(tot + 255) / 256, 256, 0, stream>>>(agg, cnt, tot);
      int cb = 61 + 44 * L + 3 * convAlpha[j];
      gemm(agg,     LD, Mp[d], Mp[d], Wlp[L][j], P(cb + 2), HID, obuf[d], HID, 1);
      gemm(hbuf[d], LD, Mp[d], Mp[d], Wrp[L][j], nullptr,   0,   obuf[d], HID, 1);
    }
    for (int k = 0; k < nDst[L]; ++k) {
      int d = actD[L][k];
      int lb = 61 + 44 * L + 30 + 2 * alphaPos[d];   // ln: b, g
      ln_relu_kernel<<<(Mn[d] + 7) / 8, 256, 0, stream>>>(obuf[d], Mn[d],
                                                          P(lb + 1), P(lb + 0));
      float* t2 = hbuf[d]; hbuf[d] = obuf[d]; obuf[d] = t2;
    }
  }

  // ---- outputs: h{bill, bill_version, committee}, stance{committee}, infl{committee} ----
  float* out = (float*)d_out;
  const size_t OFF_BV = (size_t)50000 * HID;              // 7,400,000
  const size_t OFF_CM = OFF_BV + (size_t)150000 * HID;    // 29,600,000
  const size_t OFF_ST = OFF_CM + (size_t)500 * HID;       // 29,674,000
  const size_t OFF_IN = OFF_ST + (size_t)500 * 66;        // 29,707,000

  {
    int tot = 50000 * HID;
    copy_out_kernel<<<(tot + 255) / 256, 256, 0, stream>>>(hbuf[0], out, tot);
  }
  {
    int tot = 150000 * HID;
    copy_out_kernel<<<(tot + 255) / 256, 256, 0, stream>>>(hbuf[1], out + OFF_BV, tot);
  }
  {
    int tot = 500 * HID;
    copy_out_kernel<<<(tot + 255) / 256, 256, 0, stream>>>(hbuf[4], out + OFF_CM, tot);
  }
  stance_kernel<<<(500 * 66 + 255) / 256, 256, 0, stream>>>(
      hbuf[4], P(193), P(194), out + OFF_ST, 500);
  // infl: tmp = relu(h @ W1 + b1)  (WMMA + fused ReLU), then W2/b2 + softplus
  gemm(hbuf[4], LD, 512, 512, W1p, P(47), HID, tmp, HID, 2);
  infl_kernel<<<(500 + 255) / 256, 256, 0, stream>>>(tmp, P(46), P(48),
                                                     out + OFF_IN, 500);
}